// StateLoss_69526930588391
// MI455X (gfx1250) — compile-verified
//
#include <hip/hip_runtime.h>
#include <hip/hip_bf16.h>

// ---------------- problem constants (from reference) ----------------
#define GRID_N     64
#define GRID_CELLS (GRID_N * GRID_N * GRID_N)   // 262144 floats = 1 MB
#define INV_DX     64.0f

typedef __attribute__((ext_vector_type(2))) float v2f;
typedef __attribute__((ext_vector_type(8))) float v8f;

// ---------------- kernel 1: zero diff grid + output ----------------
__global__ __launch_bounds__(256) void zero_kernel(float* __restrict__ grid,
                                                   float* __restrict__ out) {
    int i = blockIdx.x * blockDim.x + threadIdx.x;
    if (i < GRID_CELLS) grid[i] = 0.0f;
    if (i == 0) out[0] = 0.0f;
}

// ---------------- kernel 2: signed B-spline scatter ----------------
// One thread handles particle p of BOTH x (+1) and x_ref (-1) into one
// signed diff grid: 54 L2-resident global_atomic_add_f32 per thread.
__global__ __launch_bounds__(256) void scatter_kernel(const float* __restrict__ x,
                                                      const float* __restrict__ xr,
                                                      float* __restrict__ grid,
                                                      int n) {
    const int p = blockIdx.x * blockDim.x + threadIdx.x;
    if (p >= n) return;

#pragma unroll
    for (int s = 0; s < 2; ++s) {
        const float* __restrict__ src = (s == 0) ? x : xr;
        const float sign = (s == 0) ? 1.0f : -1.0f;

        const float px = src[3 * p + 0] * INV_DX;
        const float py = src[3 * p + 1] * INV_DX;
        const float pz = src[3 * p + 2] * INV_DX;

        const int bx = (int)floorf(px - 0.5f);
        const int by = (int)floorf(py - 0.5f);
        const int bz = (int)floorf(pz - 0.5f);

        const float fx = px - (float)bx;
        const float fy = py - (float)by;
        const float fz = pz - (float)bz;

        // quadratic B-spline weights per axis
        float wx[3], wy[3], wz[3];
        wx[0] = 0.5f * (1.5f - fx) * (1.5f - fx);
        wx[1] = 0.75f - (fx - 1.0f) * (fx - 1.0f);
        wx[2] = 0.5f * (fx - 0.5f) * (fx - 0.5f);
        wy[0] = 0.5f * (1.5f - fy) * (1.5f - fy);
        wy[1] = 0.75f - (fy - 1.0f) * (fy - 1.0f);
        wy[2] = 0.5f * (fy - 0.5f) * (fy - 0.5f);
        wz[0] = 0.5f * (1.5f - fz) * (1.5f - fz);
        wz[1] = 0.75f - (fz - 1.0f) * (fz - 1.0f);
        wz[2] = 0.5f * (fz - 0.5f) * (fz - 0.5f);

        // positions are in [0.05, 0.95): base in [2, 60], so stencil is in-grid
#pragma unroll
        for (int i = 0; i < 3; ++i) {
#pragma unroll
            for (int j = 0; j < 3; ++j) {
                const float wij = sign * wx[i] * wy[j];
                float* row = grid + (((bx + i) * GRID_N + (by + j)) * GRID_N + bz);
#pragma unroll
                for (int k = 0; k < 3; ++k) {
                    atomicAdd(row + k, wij * wz[k]);  // global_atomic_add_f32, L2-resident
                }
            }
        }
    }
}

// ---------------- kernel 3: WMMA abs-sum reduction ----------------
// V_WMMA_F32_16X16X4_F32 with A = ones: D[m,n] = sum_k B[k,n] + C[m,n].
// Each WMMA folds 64 fp32 |values| into the fp32 C accumulator. Rows M=0
// and M=8 of D both hold column sums, so the 32-lane sum of c[0] = 2x the
// wave total. Wave index is readfirstlane'd and the trip count is a
// compile-time constant so the loop is wave-uniform: EXEC stays all-1s
// around every WMMA (ISA requirement) with no exec-mask manipulation.
#define RED_BLOCKS 32
#define RED_WAVES  (RED_BLOCKS * (256 / 32))      // 256 waves total
#define RED_CHUNKS (GRID_CELLS / 64)              // 4096 WMMA chunks
#define RED_ITERS  (RED_CHUNKS / RED_WAVES)       // 16 per wave (exact)

__global__ __launch_bounds__(256) void reduce_abs_kernel(const float* __restrict__ grid,
                                                         float* __restrict__ out) {
    const int lane = threadIdx.x & 31;
    // scalar (SGPR) wave id -> uniform addressing, uniform loop
    const int wave = __builtin_amdgcn_readfirstlane(blockIdx.x * (256 / 32) + (threadIdx.x >> 5));

    v2f a;
    a.x = 1.0f;
    a.y = 1.0f;
    v8f c0 = {}, c1 = {};

#pragma unroll
    for (int it = 0; it < RED_ITERS; it += 2) {
        const int base0 = ((it + 0) * RED_WAVES + wave) * 64 + lane * 2;
        const int base1 = ((it + 1) * RED_WAVES + wave) * 64 + lane * 2;
        v2f b0, b1;
        b0.x = fabsf(grid[base0 + 0]);
        b0.y = fabsf(grid[base0 + 1]);
        b1.x = fabsf(grid[base1 + 0]);
        b1.y = fabsf(grid[base1 + 1]);
        // 8 args: (neg_a, A, neg_b, B, c_mod, C, reuse_a, reuse_b)
        c0 = __builtin_amdgcn_wmma_f32_16x16x4_f32(false, a, false, b0, (short)0, c0, false, false);
        c1 = __builtin_amdgcn_wmma_f32_16x16x4_f32(false, a, false, b1, (short)0, c1, false, false);
    }

    float v = c0[0] + c1[0];
#pragma unroll
    for (int off = 16; off > 0; off >>= 1) v += __shfl_xor(v, off, 32);
    if (lane == 0) atomicAdd(out, v * 0.5f);
}

// ---------------- launcher ----------------
extern "C" void kernel_launch(void* const* d_in, const int* in_sizes, int n_in,
                              void* d_out, int out_size, void* d_ws, size_t ws_size,
                              hipStream_t stream) {
    const float* x  = (const float*)d_in[0];   // [N,3] f32
    const float* xr = (const float*)d_in[1];   // [N,3] f32
    float* out  = (float*)d_out;               // scalar f32
    float* grid = (float*)d_ws;                // 64^3 f32 signed diff grid (1 MB)

    const int n = in_sizes[0] / 3;             // = 4,000,000

    // 1) zero diff grid + output
    zero_kernel<<<(GRID_CELLS + 255) / 256, 256, 0, stream>>>(grid, out);

    // 2) signed scatter of both particle sets into one grid (dominant cost:
    //    216M fp32 atomics into a 1MB L2-resident grid)
    scatter_kernel<<<(n + 255) / 256, 256, 0, stream>>>(x, xr, grid, n);

    // 3) WMMA-based |diff| sum: 256 waves x 16 fully-unrolled WMMA chunks
    reduce_abs_kernel<<<RED_BLOCKS, 256, 0, stream>>>(grid, out);
}